// CustomLoss_39204461478774
// MI455X (gfx1250) — compile-verified
//
#include <hip/hip_runtime.h>

// ---------------------------------------------------------------------------
// Change-point detection loss (false-alarm + delay-detection), MI455X/gfx1250.
//
// Memory-bound: 128 MiB in -> 1 float out; floor ~5.6us @ 23.3 TB/s.
// One 256-thread block per row. Probability row is staged into LDS with
// CDNA5 async global->LDS copies (ASYNCcnt) overlapped with the label pass,
// then consumed by a wave32 shuffle-scan for the cumprod recurrence.
// ---------------------------------------------------------------------------

#define T_DIM  2048
#define BLOCK  256
#define CHUNK  8          // T_DIM / BLOCK
#define W_WIN  8
#define ALPHA  0.5f

__global__ __launch_bounds__(BLOCK) void loss_rows_kernel(
    const float* __restrict__ outputs,
    const int*   __restrict__ labels,
    float*       __restrict__ row_loss)
{
    __shared__ float sh_p[T_DIM];        // staged probability row (8 KB)
    __shared__ float red[BLOCK];
    __shared__ int   imin[BLOCK];
    __shared__ float wprod[BLOCK / 32];

    const int row  = blockIdx.x;
    const int tid  = threadIdx.x;
    const int lane = tid & 31;
    const int wid  = tid >> 5;
    const int k0   = tid * CHUNK;

    const float* prow = outputs + (size_t)row * T_DIM;
    const int*   lrow = labels  + (size_t)row * T_DIM;

    // ---- Kick off async global->LDS copy of the probability row ----------
    // Each lane moves its 32-byte chunk with two b128 async transfers
    // (GVS mode: SGPR row base + per-lane byte offset; offset:16 applies to
    // both the LDS and global addresses). Tracked by ASYNCcnt, no VGPR data.
    {
        unsigned lds_addr = (unsigned)(unsigned long long)(&sh_p[k0]);
        unsigned goff     = (unsigned)(k0 * 4);
        asm volatile(
            "global_load_async_to_lds_b128 %0, %1, %2\n\t"
            "global_load_async_to_lds_b128 %0, %1, %2 offset:16"
            :
            : "v"(lds_addr), "v"(goff), "s"(prow)
            : "memory");
    }

    // ---- While the async copy is in flight: find ind0 from labels --------
    const int lab0 = lrow[0];
    int4 la = *(const int4*)(lrow + k0);
    int4 lb = *(const int4*)(lrow + k0 + 4);
    int loc = T_DIM;
    {
        int vv[8] = { la.x, la.y, la.z, la.w, lb.x, lb.y, lb.z, lb.w };
#pragma unroll
        for (int j = 7; j >= 0; --j)
            if (vv[j] != lab0) loc = k0 + j;   // descending -> keeps smallest
    }
    imin[tid] = loc;
    __syncthreads();
    for (int st = BLOCK / 2; st > 0; st >>= 1) {
        if (tid < st) { int o = imin[tid + st]; if (o < imin[tid]) imin[tid] = o; }
        __syncthreads();
    }
    const int  ind0raw    = imin[0];
    const bool has_change = (ind0raw < T_DIM);
    const int  ind0       = has_change ? ind0raw : 0;
    const int  L          = has_change ? ind0raw : T_DIM;

    // ---- Retire async copies, publish LDS row to all waves ---------------
    asm volatile("s_wait_asynccnt 0" ::: "memory");
    __syncthreads();

    // ---- Blocked scan of  sum_k (k+1) * p_k * prod_{j<k}(1 - p_j) --------
    // Thread t owns k in [8t, 8t+8): local partial sum s and product q.
    float s = 0.f, q = 1.f;
#pragma unroll
    for (int j = 0; j < CHUNK; ++j) {
        int   k  = k0 + j;
        float pe = (k < L) ? sh_p[k] : 0.f;   // masked tail contributes 0 / x1
        s  = fmaf((float)(k + 1) * pe, q, s);
        q *= (1.f - pe);
    }

    // wave32 inclusive product scan via shuffles
    float inc = q;
#pragma unroll
    for (int d = 1; d < 32; d <<= 1) {
        float o = __shfl_up(inc, d, 32);
        if (lane >= d) inc *= o;
    }
    float excl = __shfl_up(inc, 1, 32);
    if (lane == 0) excl = 1.f;
    if (lane == 31) wprod[wid] = inc;         // whole-wave product
    __syncthreads();
    float woff = 1.f;
    for (int w = 0; w < wid; ++w) woff *= wprod[w];
    float partial = s * (woff * excl);

    // block-wide deterministic sum
    red[tid] = partial;
    __syncthreads();
    for (int st = BLOCK / 2; st > 0; st >>= 1) {
        if (tid < st) red[tid] += red[tid + st];
        __syncthreads();
    }

    if (tid == 0) {
        float dd2 = red[0];
        float fa  = 1.f - dd2;
        float per = fa;
        if (has_change) {
            // delay-detection window p[ind0 : ind0+8] (read from staged LDS)
            int n = T_DIM - ind0; if (n > W_WIN) n = W_WIN;
            float term1 = 0.f, cpk = 1.f, pw = 0.f;
            for (int j = 0; j < n; ++j) {
                pw = sh_p[ind0 + j];
                term1 = fmaf((float)(j + 1) * pw, cpk, term1);
                if (j < n - 1) cpk *= (1.f - pw);
            }
            float dd = term1 + (float)n * (float)(W_WIN + 1) * cpk * (1.f - pw);
            per = ALPHA * dd + (1.f - ALPHA) * fa;
        }
        row_loss[row] = per;
    }
}

// Deterministic fixed-order mean over B per-row losses (single block).
__global__ __launch_bounds__(256) void reduce_mean_kernel(
    const float* __restrict__ rl, float* __restrict__ out, int B)
{
    __shared__ float red[256];
    float s = 0.f;
    for (int i = threadIdx.x; i < B; i += 256) s += rl[i];
    red[threadIdx.x] = s;
    __syncthreads();
    for (int st = 128; st > 0; st >>= 1) {
        if (threadIdx.x < st) red[threadIdx.x] += red[threadIdx.x + st];
        __syncthreads();
    }
    if (threadIdx.x == 0) out[0] = red[0] / (float)B;
}

extern "C" void kernel_launch(void* const* d_in, const int* in_sizes, int n_in,
                              void* d_out, int out_size, void* d_ws, size_t ws_size,
                              hipStream_t stream)
{
    const float* outputs = (const float*)d_in[0];
    const int*   labels  = (const int*)d_in[1];
    const int    B       = in_sizes[0] / T_DIM;

    float* row_loss = (float*)d_ws;          // B floats of scratch

    loss_rows_kernel<<<B, BLOCK, 0, stream>>>(outputs, labels, row_loss);
    reduce_mean_kernel<<<1, 256, 0, stream>>>(row_loss, (float*)d_out, B);
}